// ContrastiveLoss_64587718197396
// MI455X (gfx1250) — compile-verified
//
#include <hip/hip_runtime.h>

typedef __attribute__((ext_vector_type(2))) float v2f;
typedef __attribute__((ext_vector_type(8))) float v8f;
typedef __attribute__((ext_vector_type(4))) int   v4i;

#define N_ROWS   256
#define M_COLS   65536
#define DIM      512
#define NEG_TOPK 10
#define ROW_TILE 32            // n-rows per workgroup
#define COL_TILE 256           // m-cols per workgroup
#define K_CHUNK  32            // k staged per LDS pass
#define N_CHUNKS_K (DIM / K_CHUNK)     // 16
#define LDS_STRIDE 36          // padded k-stride (float4 aligned, conflict-free)
#define SIM_STRIDE 257         // padded sim row stride (conflict-free row scans)
#define M_CHUNKS (M_COLS / COL_TILE)   // 256
#define NEG_FILL (-1e9f)
#define EPSQ     1e-5f

#define A_ELEMS   (ROW_TILE * LDS_STRIDE)   // 1152 floats
#define B_ELEMS   (COL_TILE * LDS_STRIDE)   // 9216 floats
#define BUF_ELEMS (A_ELEMS + B_ELEMS)       // 10368 floats (>= 32*257 sim floats)

// ---- CDNA5 async global->LDS path (guarded; falls back to sync staging) ----
#if defined(__has_builtin)
#if __has_builtin(__builtin_amdgcn_global_load_async_to_lds_b128)
#define HAVE_ASYNC_LDS 1
#endif
#endif
#ifndef HAVE_ASYNC_LDS
#define HAVE_ASYNC_LDS 0
#endif

#if HAVE_ASYNC_LDS
#if __has_builtin(__builtin_amdgcn_s_wait_asynccnt)
#define WAIT_ASYNCCNT0() __builtin_amdgcn_s_wait_asynccnt(0)
#else
#define WAIT_ASYNCCNT0() asm volatile("s_wait_asynccnt 0x0" ::: "memory")
#endif
typedef __attribute__((address_space(1))) v4i* gas_v4i_t;   // prints as "v4i __device__*"
typedef __attribute__((address_space(3))) v4i* las_v4i_t;   // prints as "v4i __shared__*"
__device__ __forceinline__ void async_cp16(const float* g, float* l) {
    // per-lane 16B global -> LDS DMA, tracked with ASYNCcnt
    __builtin_amdgcn_global_load_async_to_lds_b128(
        (gas_v4i_t)(uintptr_t)g,
        (las_v4i_t)(uint32_t)(uintptr_t)l,   // low 32 bits of generic ptr = LDS offset
        0, 0);
}
#endif

// ---------------------------------------------------------------------------
// Kernel 1: 32x256 sim tile via V_WMMA_F32_16X16X4_F32 with double-buffered
// async global->LDS staging, then per-row masked pos-sums (global atomics)
// and per-(row,chunk) top-10 written to workspace.
// ---------------------------------------------------------------------------
__global__ __launch_bounds__(256)
void gemm_topk_kernel(const float* __restrict__ col,   // [N_ROWS][DIM]
                      const int*   __restrict__ labc,  // [N_ROWS]
                      const float* __restrict__ row,   // [M_COLS][DIM]
                      const int*   __restrict__ labr,  // [M_COLS]
                      float* __restrict__ ws_top,      // [N_ROWS][M_CHUNKS][NEG_TOPK]
                      float* __restrict__ g_psum,      // [N_ROWS]
                      float* __restrict__ g_pcnt)      // [N_ROWS]
{
    __shared__ __align__(16) float smem[2 * BUF_ELEMS];
    __shared__ int sLabC[ROW_TILE];
    __shared__ int sLabR[COL_TILE];

    float* sSim = smem;    // [ROW_TILE][SIM_STRIDE], aliases buffer 0 after k-loop

    const int t    = threadIdx.x;
    const int lane = t & 31;
    const int wave = t >> 5;                      // 0..7
    const int m0   = blockIdx.x * COL_TILE;
    const int n0   = blockIdx.y * ROW_TILE;

    // Wave -> 1 row-block (16 rows) x 4 col-blocks (16 cols each)
    const int rb    = wave >> 2;                  // 0..1
    const int cb0   = (wave & 3) * 4;             // 0,4,8,12
    const int lhalf = lane & 15;
    const int khi   = (lane >> 4) * 2;            // lanes 0-15 -> k+0/1, 16-31 -> k+2/3

    // per-thread staging indices (one float4 of A, eight float4 of B)
    const int rA  = t >> 3;
    const int j4A = (t & 7) * 4;

    if (t < ROW_TILE) sLabC[t] = labc[n0 + t];
    sLabR[t] = labr[m0 + t];

    v8f acc[4];
    #pragma unroll
    for (int j = 0; j < 4; ++j) acc[j] = (v8f){};

#if HAVE_ASYNC_LDS
    auto issue_chunk = [&](int kc, int buf) {
        float* sA = smem + buf * BUF_ELEMS;
        float* sB = sA + A_ELEMS;
        async_cp16(&col[(size_t)(n0 + rA) * DIM + kc + j4A], &sA[rA * LDS_STRIDE + j4A]);
        #pragma unroll
        for (int i = 0; i < 8; ++i) {
            const int lin = t + i * 256;
            const int c   = lin >> 3;
            const int j4  = (lin & 7) * 4;
            async_cp16(&row[(size_t)(m0 + c) * DIM + kc + j4], &sB[c * LDS_STRIDE + j4]);
        }
    };

    issue_chunk(0, 0);
    for (int ci = 0; ci < N_CHUNKS_K; ++ci) {
        const int buf = ci & 1;
        WAIT_ASYNCCNT0();          // my async writes for chunk ci are in LDS
        __syncthreads();           // everyone's writes visible; prev buffer free
        if (ci + 1 < N_CHUNKS_K) issue_chunk((ci + 1) * K_CHUNK, buf ^ 1);

        const float* sA = smem + buf * BUF_ELEMS;
        const float* sB = sA + A_ELEMS;
        #pragma unroll
        for (int kk = 0; kk < K_CHUNK; kk += 4) {
            const int klo = kk + khi;
            const v2f a  = *(const v2f*)&sA[(rb * 16 + lhalf) * LDS_STRIDE + klo];
            const v2f b0 = *(const v2f*)&sB[((cb0 + 0) * 16 + lhalf) * LDS_STRIDE + klo];
            const v2f b1 = *(const v2f*)&sB[((cb0 + 1) * 16 + lhalf) * LDS_STRIDE + klo];
            const v2f b2 = *(const v2f*)&sB[((cb0 + 2) * 16 + lhalf) * LDS_STRIDE + klo];
            const v2f b3 = *(const v2f*)&sB[((cb0 + 3) * 16 + lhalf) * LDS_STRIDE + klo];
            acc[0] = __builtin_amdgcn_wmma_f32_16x16x4_f32(false, a, false, b0, (short)0, acc[0], false, false);
            acc[1] = __builtin_amdgcn_wmma_f32_16x16x4_f32(false, a, false, b1, (short)0, acc[1], false, false);
            acc[2] = __builtin_amdgcn_wmma_f32_16x16x4_f32(false, a, false, b2, (short)0, acc[2], false, false);
            acc[3] = __builtin_amdgcn_wmma_f32_16x16x4_f32(false, a, false, b3, (short)0, acc[3], false, false);
        }
    }
#else
    // Fallback: synchronous staging through VGPRs
    for (int kc = 0; kc < DIM; kc += K_CHUNK) {
        float* sA = smem;
        float* sB = smem + A_ELEMS;
        __syncthreads();
        {
            const float4 v = *(const float4*)&col[(size_t)(n0 + rA) * DIM + kc + j4A];
            *(float4*)&sA[rA * LDS_STRIDE + j4A] = v;
        }
        #pragma unroll
        for (int i = 0; i < 8; ++i) {
            const int lin = t + i * 256;
            const int c   = lin >> 3;
            const int j4  = (lin & 7) * 4;
            const float4 v = *(const float4*)&row[(size_t)(m0 + c) * DIM + kc + j4];
            *(float4*)&sB[c * LDS_STRIDE + j4] = v;
        }
        __syncthreads();
        #pragma unroll
        for (int kk = 0; kk < K_CHUNK; kk += 4) {
            const int klo = kk + khi;
            const v2f a  = *(const v2f*)&sA[(rb * 16 + lhalf) * LDS_STRIDE + klo];
            const v2f b0 = *(const v2f*)&sB[((cb0 + 0) * 16 + lhalf) * LDS_STRIDE + klo];
            const v2f b1 = *(const v2f*)&sB[((cb0 + 1) * 16 + lhalf) * LDS_STRIDE + klo];
            const v2f b2 = *(const v2f*)&sB[((cb0 + 2) * 16 + lhalf) * LDS_STRIDE + klo];
            const v2f b3 = *(const v2f*)&sB[((cb0 + 3) * 16 + lhalf) * LDS_STRIDE + klo];
            acc[0] = __builtin_amdgcn_wmma_f32_16x16x4_f32(false, a, false, b0, (short)0, acc[0], false, false);
            acc[1] = __builtin_amdgcn_wmma_f32_16x16x4_f32(false, a, false, b1, (short)0, acc[1], false, false);
            acc[2] = __builtin_amdgcn_wmma_f32_16x16x4_f32(false, a, false, b2, (short)0, acc[2], false, false);
            acc[3] = __builtin_amdgcn_wmma_f32_16x16x4_f32(false, a, false, b3, (short)0, acc[3], false, false);
        }
    }
#endif

    __syncthreads();   // all waves done with buffer 0 -> safe to alias as sSim

    // C/D layout: VGPR r -> M=r (lanes 0-15) / M=r+8 (lanes 16-31), N = lane%16
    const int rAdd = (lane < 16) ? 0 : 8;
    #pragma unroll
    for (int j = 0; j < 4; ++j) {
        const int colIdx = (cb0 + j) * 16 + lhalf;
        #pragma unroll
        for (int r = 0; r < 8; ++r) {
            sSim[(rb * 16 + r + rAdd) * SIM_STRIDE + colIdx] = acc[j][r];
        }
    }
    __syncthreads();

    // One lane per row: masked pos sums + branchless sorted-insert top-10.
    if (t < ROW_TILE) {
        const int myLab = sLabC[t];
        float psum = 0.0f, pcnt = 0.0f;
        float top[NEG_TOPK];
        #pragma unroll
        for (int i = 0; i < NEG_TOPK; ++i) top[i] = NEG_FILL;

        const float* simRow = &sSim[t * SIM_STRIDE];
        for (int c = 0; c < COL_TILE; ++c) {
            float s = simRow[c];
            const bool same = (sLabR[c] == myLab);
            if (same && s < 1.0f - EPSQ) { psum += 1.0f - s; pcnt += 1.0f; }
            s = same ? NEG_FILL : s;      // mask positives out of the neg path
            #pragma unroll
            for (int i = 0; i < NEG_TOPK; ++i) {   // register insertion network
                const float hi = fmaxf(top[i], s);
                const float lo = fminf(top[i], s);
                top[i] = hi; s = lo;
            }
        }
        const int rowG = n0 + t;
        atomicAdd(&g_psum[rowG], psum);
        atomicAdd(&g_pcnt[rowG], pcnt);
        float* dst = &ws_top[((size_t)rowG * M_CHUNKS + blockIdx.x) * NEG_TOPK];
        #pragma unroll
        for (int i = 0; i < NEG_TOPK; ++i) dst[i] = top[i];
    }
}

// ---------------------------------------------------------------------------
// Kernel 2: merge per-chunk top-10 lists, combine with pos stats, reduce.
// One block, thread t owns row t.
// ---------------------------------------------------------------------------
__global__ __launch_bounds__(256)
void finalize_kernel(const float* __restrict__ ws_top,
                     const float* __restrict__ g_psum,
                     const float* __restrict__ g_pcnt,
                     float* __restrict__ out)
{
    const int t = threadIdx.x;  // row id
    float top[NEG_TOPK];
    #pragma unroll
    for (int i = 0; i < NEG_TOPK; ++i) top[i] = NEG_FILL;

    const float* src = &ws_top[(size_t)t * M_CHUNKS * NEG_TOPK];
    for (int c = 0; c < M_CHUNKS * NEG_TOPK; ++c) {
        float s = src[c];
        #pragma unroll
        for (int i = 0; i < NEG_TOPK; ++i) {
            const float hi = fmaxf(top[i], s);
            const float lo = fminf(top[i], s);
            top[i] = hi; s = lo;
        }
    }
    float tsum = 0.0f;
    #pragma unroll
    for (int i = 0; i < NEG_TOPK; ++i) tsum += top[i];
    const float neg_loss = 15.0f * (tsum / (float)NEG_TOPK);

    const float cnt = g_pcnt[t];
    const float pos_loss = (cnt > 0.0f) ? 6.0f * g_psum[t] / fmaxf(cnt, 1.0f) : 0.0f;

    __shared__ float red[256];
    red[t] = pos_loss + neg_loss;
    __syncthreads();
    for (int s2 = 128; s2 > 0; s2 >>= 1) {
        if (t < s2) red[t] += red[t + s2];
        __syncthreads();
    }
    if (t == 0) out[0] = red[0] / (float)N_ROWS;
}

// ---------------------------------------------------------------------------
extern "C" void kernel_launch(void* const* d_in, const int* in_sizes, int n_in,
                              void* d_out, int out_size, void* d_ws, size_t ws_size,
                              hipStream_t stream)
{
    const float* inputs_col  = (const float*)d_in[0];
    const int*   targets_col = (const int*)  d_in[1];
    const float* inputs_row  = (const float*)d_in[2];
    const int*   target_row  = (const int*)  d_in[3];
    float* out = (float*)d_out;

    float* ws_top = (float*)d_ws;                                // 2.62 MB
    const size_t top_elems = (size_t)N_ROWS * M_CHUNKS * NEG_TOPK;
    float* g_psum = ws_top + top_elems;
    float* g_pcnt = g_psum + N_ROWS;

    (void)hipMemsetAsync(g_psum, 0, 2 * N_ROWS * sizeof(float), stream);

    dim3 grid(M_CHUNKS, N_ROWS / ROW_TILE);   // 256 x 8
    gemm_topk_kernel<<<grid, 256, 0, stream>>>(inputs_col, targets_col,
                                               inputs_row, target_row,
                                               ws_top, g_psum, g_pcnt);
    finalize_kernel<<<1, 256, 0, stream>>>(ws_top, g_psum, g_pcnt, out);
}